// GModuleSelfAttention_17497696764421
// MI455X (gfx1250) — compile-verified
//
#include <hip/hip_runtime.h>
#include <hip/hip_bf16.h>
#include <math.h>

// ---------------------------------------------------------------------------
// Problem constants (match reference)
// ---------------------------------------------------------------------------
namespace {
constexpr int kHW    = 96 * 96;   // 9216 spatial positions
constexpr int kB     = 8;         // batch
constexpr int kCE    = 512;       // encoder channels
constexpr int kN     = 128;       // nodes per image (64 pos + 64 neg)
constexpr int kD     = 256;       // node feature dim
constexpr int kHeads = 8;
constexpr int kDh    = 32;
constexpr int kZ     = 16;        // (domain, batch) jobs: z = d*8 + b
}

typedef float v2f __attribute__((ext_vector_type(2)));
typedef float v8f __attribute__((ext_vector_type(8)));

// ---------------------------------------------------------------------------
// Generic batched fp32 WMMA GEMM:  C[z,h] = act(alpha * A[z^axor,h] @ B[z,h] + bias)
// One wave (32 lanes) computes one 16x16 output tile using V_WMMA_F32_16X16X4_F32.
// A is row-major [M,K] (lda). B is [K,N] (ldb) or, if btr, element(k,n)=B[n*ldb+k].
// K must be a multiple of 8. Two independent accumulator chains keep two
// v_wmma ops in flight (WMMA D->C chains otherwise serialize at MMA latency).
// ---------------------------------------------------------------------------
__global__ __launch_bounds__(32) void gemm_f32_wmma(
    const float* __restrict__ A, int sAz, int sAh, int lda, int a_xor,
    const float* __restrict__ B, int sBz, int sBh, int ldb, int btr,
    const float* __restrict__ bias,
    float* __restrict__ C, int sCz, int sCh, int ldc,
    int tilesN, int Kdim, float alpha, int act)
{
  const int z    = blockIdx.z;
  const int h    = blockIdx.y;
  const int tm   = blockIdx.x / tilesN;
  const int tn   = blockIdx.x % tilesN;
  const int lane = threadIdx.x;          // 0..31, wave32
  const int mi   = lane & 15;            // row within tile (A) / col within tile (B)
  const int kh   = (lane >> 4) << 1;     // lanes 16-31 hold K+2/K+3 halves

  const float* Ap   = A + (size_t)(z ^ a_xor) * sAz + (size_t)h * sAh
                        + (size_t)(tm * 16 + mi) * lda;
  const float* Bp   = B + (size_t)z * sBz + (size_t)h * sBh;
  const float* Brow = Bp + (size_t)(tn * 16 + mi) * ldb;   // btr  : contiguous in k
  const float* Bcol = Bp + (tn * 16 + mi);                 // !btr : strided by ldb

  v8f acc0 = {0.f, 0.f, 0.f, 0.f, 0.f, 0.f, 0.f, 0.f};
  v8f acc1 = {0.f, 0.f, 0.f, 0.f, 0.f, 0.f, 0.f, 0.f};

  if (btr) {
#pragma unroll 2
    for (int kb = 0; kb < Kdim; kb += 8) {
      const v2f a0 = *(const v2f*)(Ap + kb + kh);
      const v2f b0 = *(const v2f*)(Brow + kb + kh);
      const v2f a1 = *(const v2f*)(Ap + kb + 4 + kh);
      const v2f b1 = *(const v2f*)(Brow + kb + 4 + kh);
      acc0 = __builtin_amdgcn_wmma_f32_16x16x4_f32(false, a0, false, b0,
                                                   (short)0, acc0, false, false);
      acc1 = __builtin_amdgcn_wmma_f32_16x16x4_f32(false, a1, false, b1,
                                                   (short)0, acc1, false, false);
    }
  } else {
#pragma unroll 2
    for (int kb = 0; kb < Kdim; kb += 8) {
      const v2f a0 = *(const v2f*)(Ap + kb + kh);
      const v2f a1 = *(const v2f*)(Ap + kb + 4 + kh);
      v2f b0, b1;
      b0.x = Bcol[(size_t)(kb + kh) * ldb];
      b0.y = Bcol[(size_t)(kb + kh + 1) * ldb];
      b1.x = Bcol[(size_t)(kb + 4 + kh) * ldb];
      b1.y = Bcol[(size_t)(kb + 5 + kh) * ldb];
      acc0 = __builtin_amdgcn_wmma_f32_16x16x4_f32(false, a0, false, b0,
                                                   (short)0, acc0, false, false);
      acc1 = __builtin_amdgcn_wmma_f32_16x16x4_f32(false, a1, false, b1,
                                                   (short)0, acc1, false, false);
    }
  }

  const int row0 = tm * 16 + ((lane >> 4) << 3);   // D: vgpr r -> M = 8*(lane/16)+r
  const int col  = tn * 16 + mi;
  float*    Cp   = C + (size_t)z * sCz + (size_t)h * sCh;
  const float bv = bias ? bias[col] : 0.f;
#pragma unroll
  for (int r = 0; r < 8; ++r) {
    float val = (acc0[r] + acc1[r]) * alpha + bv;
    if (act) val = fmaxf(val, 0.f);
    Cp[(size_t)(row0 + r) * ldc + col] = val;
  }
}

// ---------------------------------------------------------------------------
// Top-64 / bottom-64 selection per (domain,batch). Matches jax.lax.top_k
// (monotone sigmoid skipped; ties break toward the lower index).
// ---------------------------------------------------------------------------
__global__ __launch_bounds__(256) void topk_kernel(
    const float* __restrict__ score_s, const float* __restrict__ score_t,
    int* __restrict__ idx_out)
{
  const int z = blockIdx.x;               // 0..15
  const int d = z >> 3, b = z & 7;
  const float* sp = (d == 0 ? score_s : score_t) + (size_t)b * kHW;

  __shared__ float sc[kHW];
  __shared__ float rv[256];
  __shared__ int   ri[256];
  const int t = threadIdx.x;

  for (int i = t; i < kHW; i += 256) sc[i] = sp[i];
  __syncthreads();

  int* out = idx_out + z * kN;

  // 64 positives: sequential argmax with -inf masking
  for (int it = 0; it < 64; ++it) {
    float bv = -INFINITY; int bi = 0;
    for (int i = t; i < kHW; i += 256) {
      float v = sc[i];
      if (v > bv) { bv = v; bi = i; }
    }
    rv[t] = bv; ri[t] = bi;
    __syncthreads();
    for (int s = 128; s > 0; s >>= 1) {
      if (t < s) {
        float ov = rv[t + s]; int oi = ri[t + s];
        if (ov > rv[t] || (ov == rv[t] && oi < ri[t])) { rv[t] = ov; ri[t] = oi; }
      }
      __syncthreads();
    }
    if (t == 0) { out[it] = ri[0]; sc[ri[0]] = -INFINITY; }
    __syncthreads();
  }

  // reload (reference picks negatives from the FULL array, independent of pos)
  for (int i = t; i < kHW; i += 256) sc[i] = sp[i];
  __syncthreads();

  // 64 negatives: sequential argmin with +inf masking
  for (int it = 0; it < 64; ++it) {
    float bv = INFINITY; int bi = 0;
    for (int i = t; i < kHW; i += 256) {
      float v = sc[i];
      if (v < bv) { bv = v; bi = i; }
    }
    rv[t] = bv; ri[t] = bi;
    __syncthreads();
    for (int s = 128; s > 0; s >>= 1) {
      if (t < s) {
        float ov = rv[t + s]; int oi = ri[t + s];
        if (ov < rv[t] || (ov == rv[t] && oi < ri[t])) { rv[t] = ov; ri[t] = oi; }
      }
      __syncthreads();
    }
    if (t == 0) { out[64 + it] = ri[0]; sc[ri[0]] = INFINITY; }
    __syncthreads();
  }
}

// xg[z][m][k] = feat_{d}[b][k][idx[z][m]]
__global__ __launch_bounds__(128) void gather_kernel(
    const float* __restrict__ feat_s, const float* __restrict__ feat_t,
    const int* __restrict__ idx, float* __restrict__ xg)
{
  const int node = blockIdx.x;            // z*128 + m, 0..2047
  const int z = node >> 7, m = node & 127;
  const int d = z >> 3, b = z & 7;
  const float* fp = (d == 0 ? feat_s : feat_t) + (size_t)b * kCE * kHW;
  const int pix = idx[z * kN + m];
  float* op = xg + (size_t)node * kCE;
  for (int k = threadIdx.x; k < kCE; k += 128)
    op[k] = fp[(size_t)k * kHW + pix];
}

// In-place LayerNorm over rows of 256 (elementwise_affine=False), optional ReLU
__global__ __launch_bounds__(256) void ln_kernel(float* __restrict__ x, int relu)
{
  float* p = x + (size_t)blockIdx.x * kD;
  const int t = threadIdx.x;
  const float v = p[t];
  __shared__ float s1[256], s2[256];
  s1[t] = v; s2[t] = v * v;
  __syncthreads();
  for (int s = 128; s > 0; s >>= 1) {
    if (t < s) { s1[t] += s1[t + s]; s2[t] += s2[t + s]; }
    __syncthreads();
  }
  const float mean = s1[0] * (1.f / kD);
  const float var  = s2[0] * (1.f / kD) - mean * mean;
  float r = (v - mean) * rsqrtf(var + 1e-5f);
  if (relu) r = fmaxf(r, 0.f);
  p[t] = r;
}

// In-place softmax over contiguous rows of 128
__global__ __launch_bounds__(128) void softmax128(float* __restrict__ att)
{
  float* p = att + (size_t)blockIdx.x * 128;
  const int t = threadIdx.x;
  __shared__ float sm[128];
  const float v = p[t];
  sm[t] = v; __syncthreads();
  for (int s = 64; s > 0; s >>= 1) { if (t < s) sm[t] = fmaxf(sm[t], sm[t + s]); __syncthreads(); }
  const float mx = sm[0];
  __syncthreads();
  const float e = expf(v - mx);
  sm[t] = e; __syncthreads();
  for (int s = 64; s > 0; s >>= 1) { if (t < s) sm[t] += sm[t + s]; __syncthreads(); }
  p[t] = e / sm[0];
}

// InstanceNorm2d(1) over each [128,128] affinity matrix, in place
__global__ __launch_bounds__(256) void inorm_kernel(float* __restrict__ M)
{
  float* p = M + (size_t)blockIdx.x * kN * kN;
  const int t = threadIdx.x;
  __shared__ float s1[256], s2[256];
  float a = 0.f, q = 0.f;
  for (int i = t; i < kN * kN; i += 256) { float v = p[i]; a += v; q += v * v; }
  s1[t] = a; s2[t] = q;
  __syncthreads();
  for (int s = 128; s > 0; s >>= 1) {
    if (t < s) { s1[t] += s1[t + s]; s2[t] += s2[t + s]; }
    __syncthreads();
  }
  const float mean = s1[0] * (1.f / (kN * kN));
  const float var  = s2[0] * (1.f / (kN * kN)) - mean * mean;
  const float inv  = rsqrtf(var + 1e-5f);
  for (int i = t; i < kN * kN; i += 256) p[i] = (p[i] - mean) * inv;
}

// K -= logsumexp over last dim (rows of 128, contiguous)
__global__ __launch_bounds__(128) void row_lse_kernel(float* __restrict__ Km)
{
  float* p = Km + (size_t)blockIdx.x * 128;
  const int t = threadIdx.x;
  __shared__ float sm[128];
  const float v = p[t];
  sm[t] = v; __syncthreads();
  for (int s = 64; s > 0; s >>= 1) { if (t < s) sm[t] = fmaxf(sm[t], sm[t + s]); __syncthreads(); }
  const float mx = sm[0];
  __syncthreads();
  sm[t] = expf(v - mx); __syncthreads();
  for (int s = 64; s > 0; s >>= 1) { if (t < s) sm[t] += sm[t + s]; __syncthreads(); }
  p[t] = v - (mx + logf(sm[0]));
}

// K -= logsumexp over dim 1 (columns, stride-128)
__global__ __launch_bounds__(128) void col_lse_kernel(float* __restrict__ Km)
{
  const int b = blockIdx.x >> 7, c = blockIdx.x & 127;
  float* p = Km + (size_t)b * kN * kN + c;
  const int t = threadIdx.x;
  __shared__ float sm[128];
  const float v = p[(size_t)t * 128];
  sm[t] = v; __syncthreads();
  for (int s = 64; s > 0; s >>= 1) { if (t < s) sm[t] = fmaxf(sm[t], sm[t + s]); __syncthreads(); }
  const float mx = sm[0];
  __syncthreads();
  sm[t] = expf(v - mx); __syncthreads();
  for (int s = 64; s > 0; s >>= 1) { if (t < s) sm[t] += sm[t + s]; __syncthreads(); }
  p[(size_t)t * 128] = v - (mx + logf(sm[0]));
}

// Per-z BCEWithLogits partial (second cls layer fused: 128-dot + bias)
__global__ __launch_bounds__(128) void node_loss_kernel(
    const float* __restrict__ hid, const float* __restrict__ W2,
    const float* __restrict__ b2, float* __restrict__ partial)
{
  const int z = blockIdx.x, m = threadIdx.x;
  const float* hp = hid + ((size_t)z * kN + m) * 128;
  float l = b2[0];
  for (int j = 0; j < 128; ++j) l += hp[j] * W2[j];
  const float lab  = (m < 64) ? 1.f : 0.f;
  const float loss = fmaxf(l, 0.f) - l * lab + log1pf(expf(-fabsf(l)));
  __shared__ float sm[128];
  sm[m] = loss; __syncthreads();
  for (int s = 64; s > 0; s >>= 1) { if (m < s) sm[m] += sm[m + s]; __syncthreads(); }
  if (m == 0) partial[z] = sm[0] * (1.f / (kB * kN));   // each BCE mean divides by 1024
}

// Focal BCE partial per batch: P = exp(K), gt = (row<64)==(col<64)
__global__ __launch_bounds__(256) void mat_loss_kernel(
    const float* __restrict__ Km, float* __restrict__ partial)
{
  const float* p = Km + (size_t)blockIdx.x * kN * kN;
  const int t = threadIdx.x;
  float acc = 0.f;
  for (int i = t; i < kN * kN; i += 256) {
    const int r = i >> 7, c = i & 127;
    float pt = expf(p[i]);
    pt = fminf(fmaxf(pt, 1e-6f), 1.f - 1e-6f);
    const float gt = ((r < 64) == (c < 64)) ? 1.f : 0.f;
    const float om = 1.f - pt;
    acc += -0.25f * om * om * gt * logf(pt) - 0.75f * pt * pt * (1.f - gt) * logf(om);
  }
  __shared__ float sm[256];
  sm[t] = acc; __syncthreads();
  for (int s = 128; s > 0; s >>= 1) { if (t < s) sm[t] += sm[t + s]; __syncthreads(); }
  if (t == 0) partial[blockIdx.x] = sm[0];
}

__global__ void finalize_kernel(const float* __restrict__ partial, float* __restrict__ out_loss)
{
  if (threadIdx.x == 0 && blockIdx.x == 0) {
    float nl = 0.f;
    for (int i = 0; i < 16; ++i) nl += partial[i];
    float ml = 0.f;
    for (int i = 0; i < 8; ++i) ml += partial[16 + i];
    out_loss[0] = nl + 10.f * (ml * (1.f / (kB * kN * kN)));
  }
}

__global__ __launch_bounds__(256) void copy_kernel(
    const float* __restrict__ src, float* __restrict__ dst, int n)
{
  const int i = blockIdx.x * 256 + threadIdx.x;
  if (i < n) dst[i] = src[i];
}

// ---------------------------------------------------------------------------
// Host-side orchestration
// ---------------------------------------------------------------------------
static inline void gemm(hipStream_t stream,
                        const float* A, int sAz, int sAh, int lda, int a_xor,
                        const float* B, int sBz, int sBh, int ldb, int btr,
                        const float* bias, float* C, int sCz, int sCh, int ldc,
                        int M, int N, int Kd, int nh, int nz, float alpha, int act)
{
  dim3 grid((M / 16) * (N / 16), nh, nz);
  gemm_f32_wmma<<<grid, dim3(32), 0, stream>>>(
      A, sAz, sAh, lda, a_xor, B, sBz, sBh, ldb, btr,
      bias, C, sCz, sCh, ldc, N / 16, Kd, alpha, act);
}

extern "C" void kernel_launch(void* const* d_in, const int* in_sizes, int n_in,
                              void* d_out, int out_size, void* d_ws, size_t ws_size,
                              hipStream_t stream)
{
  (void)in_sizes; (void)n_in; (void)out_size; (void)ws_size;

  const float* feat_s  = (const float*)d_in[0];
  const float* score_s = (const float*)d_in[1];
  const float* feat_t  = (const float*)d_in[2];
  const float* score_t = (const float*)d_in[3];
  const float* head_W1 = (const float*)d_in[4];   // [512,256]
  const float* head_b1 = (const float*)d_in[5];
  const float* head_W2 = (const float*)d_in[6];   // [256,256]
  const float* head_b2 = (const float*)d_in[7];
  const float* intra_W = (const float*)d_in[8];   // [4,256,256]
  const float* intra_b = (const float*)d_in[9];   // [4,256]
  const float* cross_W = (const float*)d_in[10];
  const float* cross_b = (const float*)d_in[11];
  const float* cls_W1  = (const float*)d_in[12];  // [256,128]
  const float* cls_b1  = (const float*)d_in[13];
  const float* cls_W2  = (const float*)d_in[14];  // [128,1]
  const float* cls_b2  = (const float*)d_in[15];
  const float* aff_A   = (const float*)d_in[16];  // [256,256]

  float* out = (float*)d_out;                     // [8*128*256] n1, then [1] loss

  // ---- workspace layout (floats) ----
  const int ND = kN * kD;                         // 32768 floats per (z) node block
  int*   idxb    = (int*)d_ws;                    // 16*128 ints
  float* ws      = (float*)d_ws;
  float* partial = ws + 2048;                     // 24 floats
  float* xg      = ws + 4096;                     // 16*128*512  (also reused: cls hidden)
  float* nA      = xg  + (size_t)kZ * kN * kCE;   // 16*128*256
  float* nB      = nA  + (size_t)kZ * ND;
  float* qb      = nB  + (size_t)kZ * ND;         // (also reused: n1 @ aff_A)
  float* kb      = qb  + (size_t)kZ * ND;         // (also reused: sinkhorn K matrix)
  float* vb      = kb  + (size_t)kZ * ND;
  float* ctx     = vb  + (size_t)kZ * ND;
  float* att     = ctx + (size_t)kZ * ND;         // 16*8*128*128

  const float inv_sqrt_dh = 0.17677669529663688f; // 1/sqrt(32)
  const int sAtt_z = kHeads * kN * kN;            // 131072
  const int sAtt_h = kN * kN;                     // 16384

  // 1) sample points (top-64 / bottom-64 per image, both domains)
  topk_kernel<<<dim3(kZ), dim3(256), 0, stream>>>(score_s, score_t, idxb);
  // 2) gather node features [16][128][512]
  gather_kernel<<<dim3(kZ * kN), dim3(128), 0, stream>>>(feat_s, feat_t, idxb, xg);

  // 3) head: LN(ReLU(LN(x@W1+b1)) @ W2 + b2)
  gemm(stream, xg, kN * kCE, 0, kCE, 0, head_W1, 0, 0, kD, 0, head_b1,
       nA, ND, 0, kD, kN, kD, kCE, 1, kZ, 1.f, 0);
  ln_kernel<<<dim3(kZ * kN), dim3(256), 0, stream>>>(nA, 1);
  gemm(stream, nA, ND, 0, kD, 0, head_W2, 0, 0, kD, 0, head_b2,
       nB, ND, 0, kD, kN, kD, kD, 1, kZ, 1.f, 0);
  ln_kernel<<<dim3(kZ * kN), dim3(256), 0, stream>>>(nB, 0);

  // 4) intra-domain self-attention (shared weights): nB -> nA
  gemm(stream, nB, ND, 0, kD, 0, intra_W + 0 * kD * kD, 0, 0, kD, 0, intra_b + 0 * kD,
       qb, ND, 0, kD, kN, kD, kD, 1, kZ, 1.f, 0);
  gemm(stream, nB, ND, 0, kD, 0, intra_W + 1 * kD * kD, 0, 0, kD, 0, intra_b + 1 * kD,
       kb, ND, 0, kD, kN, kD, kD, 1, kZ, 1.f, 0);
  gemm(stream, nB, ND, 0, kD, 0, intra_W + 2 * kD * kD, 0, 0, kD, 0, intra_b + 2 * kD,
       vb, ND, 0, kD, kN, kD, kD, 1, kZ, 1.f, 0);
  gemm(stream, qb, ND, kDh, kD, 0, kb, ND, kDh, kD, 1, nullptr,
       att, sAtt_z, sAtt_h, kN, kN, kN, kDh, kHeads, kZ, inv_sqrt_dh, 0);
  softmax128<<<dim3(kZ * kHeads * kN), dim3(128), 0, stream>>>(att);
  gemm(stream, att, sAtt_z, sAtt_h, kN, 0, vb, ND, kDh, kD, 0, nullptr,
       ctx, ND, kDh, kD, kN, kDh, kN, kHeads, kZ, 1.f, 0);
  gemm(stream, ctx, ND, 0, kD, 0, intra_W + 3 * kD * kD, 0, 0, kD, 0, intra_b + 3 * kD,
       nA, ND, 0, kD, kN, kD, kD, 1, kZ, 1.f, 0);

  // 5) cross-domain attention (q from z, k/v from z^8): nA -> nB
  gemm(stream, nA, ND, 0, kD, 0, cross_W + 0 * kD * kD, 0, 0, kD, 0, cross_b + 0 * kD,
       qb, ND, 0, kD, kN, kD, kD, 1, kZ, 1.f, 0);
  gemm(stream, nA, ND, 0, kD, 8, cross_W + 1 * kD * kD, 0, 0, kD, 0, cross_b + 1 * kD,
       kb, ND, 0, kD, kN, kD, kD, 1, kZ, 1.f, 0);
  gemm(stream, nA, ND, 0, kD, 8, cross_W + 2 * kD * kD, 0, 0, kD, 0, cross_b + 2 * kD,
       vb, ND, 0, kD, kN, kD, kD, 1, kZ, 1.f, 0);
  gemm(stream, qb, ND, kDh, kD, 0, kb, ND, kDh, kD, 1, nullptr,
       att, sAtt_z, sAtt_h, kN, kN, kN, kDh, kHeads, kZ, inv_sqrt_dh, 0);
  softmax128<<<dim3(kZ * kHeads * kN), dim3(128), 0, stream>>>(att);
  gemm(stream, att, sAtt_z, sAtt_h, kN, 0, vb, ND, kDh, kD, 0, nullptr,
       ctx, ND, kDh, kD, kN, kDh, kN, kHeads, kZ, 1.f, 0);
  gemm(stream, ctx, ND, 0, kD, 0, cross_W + 3 * kD * kD, 0, 0, kD, 0, cross_b + 3 * kD,
       nB, ND, 0, kD, kN, kD, kD, 1, kZ, 1.f, 0);

  // 6) output n1 = cross result for domain 0 (z = 0..7)
  copy_kernel<<<dim3((kB * ND + 255) / 256), dim3(256), 0, stream>>>(nB, out, kB * ND);

  // 7) node classification loss (cls hidden reuses xg)
  gemm(stream, nB, ND, 0, kD, 0, cls_W1, 0, 0, 128, 0, cls_b1,
       xg, kN * 128, 0, 128, kN, 128, kD, 1, kZ, 1.f, 1);
  node_loss_kernel<<<dim3(kZ), dim3(128), 0, stream>>>(xg, cls_W2, cls_b2, partial);

  // 8) affinity: Mmat = (n1 @ aff_A) @ n2^T   (T reuses qb, Mmat reuses kb)
  gemm(stream, nB, ND, 0, kD, 0, aff_A, 0, 0, kD, 0, nullptr,
       qb, ND, 0, kD, kN, kD, kD, 1, kB, 1.f, 0);
  gemm(stream, qb, ND, 0, kD, 0, nB + (size_t)kB * ND, ND, 0, kD, 1, nullptr,
       kb, kN * kN, 0, kN, kN, kN, kD, 1, kB, 1.f, 0);

  // 9) instance norm + 10 sinkhorn iterations (log space)
  inorm_kernel<<<dim3(kB), dim3(256), 0, stream>>>(kb);
  for (int it = 0; it < 10; ++it) {
    row_lse_kernel<<<dim3(kB * kN), dim3(128), 0, stream>>>(kb);
    col_lse_kernel<<<dim3(kB * kN), dim3(128), 0, stream>>>(kb);
  }

  // 10) focal matching loss + total
  mat_loss_kernel<<<dim3(kB), dim3(256), 0, stream>>>(kb, partial + 16);
  finalize_kernel<<<dim3(1), dim3(32), 0, stream>>>(partial, out + (size_t)kB * ND);
}